// MyLSTM_44581760532734
// MI455X (gfx1250) — compile-verified
//
#include <hip/hip_runtime.h>
#include <hip/hip_bf16.h>
#include <math.h>

// Problem constants (match reference): T=256, B=512, IN=512, H=512, L=2
#define TT   256
#define BB   512
#define INN  512
#define HH   512
#define NG   2048                    // 4*H, gate order i,f,g,o
#define BH   (BB * HH)               // elements per state buffer
#define MS   ((size_t)INN * NG)      // elements per weight matrix (512*2048)
#define KBLK 256                     // K per staging phase
#define LDSROW 260                   // padded row stride (floats): 4-bank shift/row

typedef __attribute__((ext_vector_type(2))) float v2f;
typedef __attribute__((ext_vector_type(4))) float v4f;
typedef __attribute__((ext_vector_type(8))) float v8f;
typedef __attribute__((ext_vector_type(4))) int   v4i;

__device__ __forceinline__ float sigmoidf_(float x) {
    return 1.0f / (1.0f + __expf(-x));
}
__device__ __forceinline__ float tanhf_(float x) {
    return 2.0f / (1.0f + __expf(-2.0f * x)) - 1.0f;
}

// ---- CDNA5 async global->LDS staging (ASYNCcnt path), with safe fallback ----
#if defined(__AMDGCN__) && __has_builtin(__builtin_amdgcn_global_load_async_to_lds_b128)
#define HAVE_ASYNC_LDS 1
// Builtin signature (from hipcc diagnostics): param 0 is a NON-const AS1
// (global, printed "__device__") pointer to a 16-byte int vector, param 1 the
// AS3 LDS destination, then imm offset + imm cpol.
typedef __attribute__((address_space(1))) v4i* g_v4i_ptr;
typedef __attribute__((address_space(3))) v4i* l_v4i_ptr;
__device__ __forceinline__ void async_copy16(const float* gp, float* lp) {
    __builtin_amdgcn_global_load_async_to_lds_b128(
        (g_v4i_ptr)(uintptr_t)gp, (l_v4i_ptr)(uintptr_t)lp, 0, 0);
}
__device__ __forceinline__ void wait_async0() {
#if __has_builtin(__builtin_amdgcn_s_wait_asynccnt)
    __builtin_amdgcn_s_wait_asynccnt(0);
#else
    asm volatile("s_wait_asynccnt 0x0" ::: "memory");
#endif
}
#else
#define HAVE_ASYNC_LDS 0
__device__ __forceinline__ void async_copy16(const float* gp, float* lp) {
    *(v4f*)lp = *(const v4f*)gp;     // global_load_b128 + ds_store_b128
}
__device__ __forceinline__ void wait_async0() {}
#endif

// One K-block of the GEMM: A fragments from the LDS-staged tile, B fragments
// from the pair-interleaved weights (one aligned b64 per gate), accumulated
// with V_WMMA_F32_16X16X4_F32 (full f32 precision on the matrix pipe).
__device__ __forceinline__ void gemm_kblock(const float* __restrict__ sA,   // [32][LDSROW]
                                            const float* __restrict__ Wi,  // this matrix
                                            int kbase, int half, int lm, int nCol,
                                            v8f acc[2][4]) {
    #pragma unroll 2
    for (int kk = 0; kk < KBLK; kk += 4) {
        const int ka = kk + 2 * half;               // lane-half K offset in block
        const int kp = ((kbase + kk) >> 1) + half;  // global pair index = (kbase+ka)>>1
        v2f a0 = *(const v2f*)(sA + lm * LDSROW + ka);          // ds_load_b64
        v2f a1 = *(const v2f*)(sA + (16 + lm) * LDSROW + ka);   // ds_load_b64
        const float* wk = Wi + ((size_t)kp * NG + nCol) * 2;
        #pragma unroll
        for (int g = 0; g < 4; ++g) {
            v2f b = *(const v2f*)(wk + g * 1024);   // {W[ka,n], W[ka+1,n]}, gate g
            acc[0][g] = __builtin_amdgcn_wmma_f32_16x16x4_f32(
                false, a0, false, b, (short)0, acc[0][g], false, false);
            acc[1][g] = __builtin_amdgcn_wmma_f32_16x16x4_f32(
                false, a1, false, b, (short)0, acc[1][g], false, false);
        }
    }
}

// Diagonal-fused step: blockIdx.z==0 runs layer 0 at t=s, blockIdx.z==1 runs
// layer 1 at t=s-1 (independent -> one launch, 257 launches total).
// Block = 256 threads = 8 waves tiling N; each wave owns a 32(M)x16(N) tile
// across all four gates. A tiles are staged cooperatively into LDS (async),
// removing the 8x cross-wave duplication of input-row reads.
// Grid: (BB/32, HH/128, 2).
__global__ __launch_bounds__(256) void lstm_diag_kernel(
    const float* __restrict__ x0,    // X + t*BB*INN (layer-0 input)
    const float* __restrict__ h0rd,  // h0 parity s&1   (L0 recurrent in, L1 feed in)
    float*       __restrict__ h0wr,  // h0 parity (s+1)&1 (L0 hidden out)
    float*       __restrict__ c0,    // layer-0 cell state (in-place)
    const float* __restrict__ h1rd,  // h1 parity (s+1)&1 (L1 recurrent in)
    float*       __restrict__ h1wr,  // h1 parity s&1     (L1 hidden out)
    float*       __restrict__ c1,    // layer-1 cell state (in-place)
    float*       __restrict__ y1,    // d_out + (s-1)*BB*HH
    const float* __restrict__ Wi,    // 4 interleaved matrices: WxI0,WhI0,WxI1,WhI1
    const float* __restrict__ bsum,  // [2][2048] = bx+bh per layer
    int doL0, int doL1)
{
    __shared__ float sA[32 * LDSROW];                 // 33280 B staged A tile

    const int role = blockIdx.z;
    if (role == 0) { if (!doL0) return; } else { if (!doL1) return; }

    const float* xin  = role ? h0rd : x0;
    const float* hin  = role ? h1rd : h0rd;
    float*       cbuf = role ? c1   : c0;
    float*       hout = role ? h1wr : h0wr;
    float*       yout = role ? y1   : nullptr;
    const float* W    = Wi + (size_t)role * 2 * MS;   // [WxI | WhI] this layer
    const float* bs   = bsum + role * NG;

    const int lane = threadIdx.x & 31;
    const int wave = threadIdx.x >> 5;
    const int half = lane >> 4;
    const int lm   = lane & 15;

    const int m0   = blockIdx.x * 32;                 // batch-tile base (32 rows)
    const int nCol = blockIdx.y * 128 + wave * 16 + lm;  // per-gate column

    v8f acc[2][4] = {};                               // [M-subtile][gate i,f,g,o]

    const float* abase[2] = { xin + (size_t)m0 * 512, hin + (size_t)m0 * 512 };

    // 4 phases: {x,h} x {K 0:256, 256:512}. Stage 32KB A block, then 64 k-steps.
    #pragma unroll 1
    for (int phase = 0; phase < 4; ++phase) {
        const int mat   = phase >> 1;                 // 0: x@Wx, 1: h@Wh
        const int kbase = (phase & 1) * KBLK;
        const float* src = abase[mat] + kbase;

        // Cooperative stage: 2048 x b128 chunks over 256 threads (8 each).
        #pragma unroll
        for (int i = 0; i < 8; ++i) {
            const int chunk = (int)threadIdx.x + i * 256;
            const int row   = chunk >> 6;             // 0..31
            const int c4    = (chunk & 63) * 4;       // 0..252, 16B granules
            async_copy16(src + (size_t)row * 512 + c4, &sA[row * LDSROW + c4]);
        }
        wait_async0();
        __syncthreads();

        gemm_kblock(sA, W + (size_t)mat * MS, kbase, half, lm, nCol, acc);
        __syncthreads();                              // before next phase overwrites sA
    }

    const float bi = bs[0 * 512 + nCol];
    const float bf = bs[1 * 512 + nCol];
    const float bg = bs[2 * 512 + nCol];
    const float bo = bs[3 * 512 + nCol];

    // C/D layout: VGPR r holds M = base + 8*half + r, N = nCol
    #pragma unroll
    for (int mt = 0; mt < 2; ++mt) {
        const int mRow = m0 + mt * 16 + 8 * half;
        #pragma unroll
        for (int r = 0; r < 8; ++r) {
            const size_t idx = (size_t)(mRow + r) * 512 + nCol;
            const float gi = sigmoidf_(acc[mt][0][r] + bi);
            const float gf = sigmoidf_(acc[mt][1][r] + bf);
            const float gg = tanhf_  (acc[mt][2][r] + bg);
            const float go = sigmoidf_(acc[mt][3][r] + bo);
            const float cOld = cbuf[idx];
            const float cNew = gf * cOld + gi * gg;
            const float hNew = go * tanhf_(cNew);
            cbuf[idx] = cNew;
            hout[idx] = hNew;
            if (yout) yout[idx] = hNew;
        }
    }
}

__global__ void zero_ws_kernel(float* __restrict__ p, int n) {
    int i = blockIdx.x * blockDim.x + threadIdx.x;
    if (i < n) p[i] = 0.0f;
}

__global__ void bias_kernel(const float* __restrict__ bx,
                            const float* __restrict__ bh,
                            float* __restrict__ bsum) {
    int i = blockIdx.x * blockDim.x + threadIdx.x;
    if (i < 2 * NG) bsum[i] = bx[i] + bh[i];
}

// Pair-interleave weights: dst[m][((k>>1)*NG + n)*2 + (k&1)] = src_m[k*NG + n]
// Matrix order m: 0=Wx[0], 1=Wh[0], 2=Wx[1], 3=Wh[1].
__global__ void interleave_kernel(const float* __restrict__ Wx,
                                  const float* __restrict__ Wh,
                                  float* __restrict__ Wi) {
    size_t gid = (size_t)blockIdx.x * blockDim.x + threadIdx.x;
    if (gid >= 4 * MS) return;
    const int m = (int)(gid / MS);
    const size_t e = gid % MS;
    const int k = (int)(e / NG);
    const int n = (int)(e % NG);
    const int l = m >> 1;
    const float* src = (m & 1) ? (Wh + (size_t)l * MS) : (Wx + (size_t)l * MS);
    Wi[(size_t)m * MS + ((size_t)(k >> 1) * NG + n) * 2 + (k & 1)] = src[e];
}

extern "C" void kernel_launch(void* const* d_in, const int* in_sizes, int n_in,
                              void* d_out, int out_size, void* d_ws, size_t ws_size,
                              hipStream_t stream) {
    const float* X  = (const float*)d_in[0];   // [T, B, IN]
    const float* Wx = (const float*)d_in[1];   // [L, IN, 4H]
    const float* Wh = (const float*)d_in[2];   // [L, H, 4H]
    const float* bx = (const float*)d_in[3];   // [L, 4H]
    const float* bh = (const float*)d_in[4];   // [L, 4H]
    float* out = (float*)d_out;                // [T, B, H]

    // Workspace (~23 MB): states | interleaved weights | summed biases
    float* w = (float*)d_ws;
    float* h0[2] = { w + 0 * BH, w + 1 * BH };
    float* h1[2] = { w + 2 * BH, w + 3 * BH };
    float* c0   = w + 4 * BH;
    float* c1   = w + 5 * BH;
    float* Wi   = w + 6 * BH;                  // 4 * MS floats (16 MB)
    float* bsum = Wi + 4 * MS;                 // 2 * NG floats

    // Per-call prep (ws is poisoned by the harness).
    zero_ws_kernel<<<(6 * BH + 255) / 256, 256, 0, stream>>>(w, 6 * BH);
    bias_kernel<<<(2 * NG + 255) / 256, 256, 0, stream>>>(bx, bh, bsum);
    interleave_kernel<<<(unsigned)((4 * MS + 255) / 256), 256, 0, stream>>>(Wx, Wh, Wi);

    dim3 grid(BB / 32, HH / 128, 2);           // (16, 4, 2)
    dim3 block(256);

    // Diagonal pipeline: launch s = { layer0 @ t=s, layer1 @ t=s-1 }.
    for (int s = 0; s <= TT; ++s) {
        const float* x0 = X + (size_t)(s < TT ? s : TT - 1) * BB * INN;
        float* h0rd = h0[s & 1];
        float* h0wr = h0[(s + 1) & 1];
        float* h1rd = h1[(s + 1) & 1];         // == parity (s-1)&1
        float* h1wr = h1[s & 1];
        float* y1   = (s >= 1) ? out + (size_t)(s - 1) * BB * HH : out;
        lstm_diag_kernel<<<grid, block, 0, stream>>>(
            x0, h0rd, h0wr, c0, h1rd, h1wr, c1, y1,
            Wi, bsum, (int)(s < TT), (int)(s >= 1));
    }
}